// ParallelMLPSlow_57612691308943
// MI455X (gfx1250) — compile-verified
//
#include <hip/hip_runtime.h>
#include <hip/hip_bf16.h>

// MI455X / gfx1250, wave32. fp32 MLP bank of K=64 nets over shared batch.
// One workgroup per (k, 64-wide batch tile). Activations [512 x 64] in LDS,
// stored row-PAIR interleaved: addr(k,b) = (k>>1)*PAIRW + 2*b + (k&1).
//  -> WMMA B fragments are single ds_load_b64 (reg-pair ready, no shuffles)
//  -> ReLU store phase writes row pairs as ds_store_b64
//  -> PAIRW=152: B-load half-wave bank shift 24, store shift 32 (disjoint)
// GEMMs via V_WMMA_F32_16X16X4_F32 (fp32-exact vs reference).

#define H_DIM   512
#define D_IN    128
#define BTILE   64
#define PAIRW   152         // dword stride per row-pair (64 cols * 2 + 24 pad)
#define NUM_K   64
#define BATCH   2048

typedef __attribute__((ext_vector_type(2))) float v2f;
typedef __attribute__((ext_vector_type(8))) float v8f;

// One hidden layer: hbuf <- relu(W[H x Kdim] @ hbuf[0..Kdim-1][0..63] + bias)
// Each wave computes rows [wave*64, wave*64+63] (4 row-tiles x 4 col-tiles).
__device__ __forceinline__ void run_layer(const float* __restrict__ W,
                                          const float* __restrict__ bias,
                                          int Kdim, int ldw,
                                          float* __restrict__ hbuf,
                                          int wave, int lane)
{
    const int lhalf = lane >> 4;     // 0: lanes 0-15, 1: lanes 16-31
    const int l16   = lane & 15;
    const int row0  = wave * 64;

    v8f acc[4][4];
    #pragma unroll
    for (int rt = 0; rt < 4; ++rt)
        #pragma unroll
        for (int ct = 0; ct < 4; ++ct)
            acc[rt][ct] = (v8f){0.f,0.f,0.f,0.f,0.f,0.f,0.f,0.f};

    #pragma unroll 2
    for (int k0 = 0; k0 < Kdim; k0 += 4) {
        // B fragments (4x16, KxN): rows (k0+lhalf*2, k0+lhalf*2+1) are one
        // LDS pair -> single aligned b64 load per col-tile.
        const int pbase = (k0 >> 1) + lhalf;      // pair index
        v2f bfrag[4];
        #pragma unroll
        for (int ct = 0; ct < 4; ++ct) {
            const int col = ct * 16 + l16;
            bfrag[ct] = *(const v2f*)(hbuf + (size_t)pbase * PAIRW + 2 * col);
        }
        // A fragments (16x4, MxK) from row-major W:
        //   VGPR0[lane] = A[M=l16][k0 + lhalf*2], VGPR1 = next K column
        v2f afrag[4];
        #pragma unroll
        for (int rt = 0; rt < 4; ++rt) {
            const int row = row0 + rt * 16 + l16;
            afrag[rt] = *(const v2f*)(W + (size_t)row * ldw + k0 + lhalf * 2);
        }
        #pragma unroll
        for (int rt = 0; rt < 4; ++rt)
            #pragma unroll
            for (int ct = 0; ct < 4; ++ct)
                acc[rt][ct] = __builtin_amdgcn_wmma_f32_16x16x4_f32(
                    false, afrag[rt], false, bfrag[ct],
                    (short)0, acc[rt][ct], false, false);
    }

    __syncthreads();   // all waves done reading hbuf before overwrite

    // D layout: VGPR r -> lanes 0-15: (M=rowb+r, N=l16); lanes 16-31: M+8.
    // rowb is even and r steps by 2 -> rows (r, r+1) form one LDS pair:
    // write both as a single b64 store.
    #pragma unroll
    for (int rt = 0; rt < 4; ++rt) {
        const int rowb = row0 + rt * 16 + lhalf * 8;
        #pragma unroll
        for (int r = 0; r < 8; r += 2) {
            const float bv0 = bias[rowb + r];
            const float bv1 = bias[rowb + r + 1];
            #pragma unroll
            for (int ct = 0; ct < 4; ++ct) {
                float v0 = acc[rt][ct][r]     + bv0;
                float v1 = acc[rt][ct][r + 1] + bv1;
                v2f v;
                v.x = v0 > 0.f ? v0 : 0.f;
                v.y = v1 > 0.f ? v1 : 0.f;
                *(v2f*)(hbuf + (size_t)((rowb + r) >> 1) * PAIRW
                              + 2 * (ct * 16 + l16)) = v;
            }
        }
    }
    __syncthreads();
}

__global__ __launch_bounds__(256)
void parallel_mlp_kernel(const float* __restrict__ x,
                         const float* __restrict__ W1, const float* __restrict__ b1,
                         const float* __restrict__ W2, const float* __restrict__ b2,
                         const float* __restrict__ W3, const float* __restrict__ b3,
                         const float* __restrict__ W4, const float* __restrict__ b4,
                         float* __restrict__ out)
{
    __shared__ float hbuf[(H_DIM / 2) * PAIRW];   // 152 KB activation tile
    __shared__ float red[256];

    const int k    = blockIdx.y;           // which MLP copy
    const int b0   = blockIdx.x * BTILE;   // batch tile start
    const int tid  = threadIdx.x;
    const int wave = tid >> 5;
    const int lane = tid & 31;

    // Stage x tile transposed into pair layout: h[d][b] = x[b0+b][d].
    // Each thread moves a (d, d+1) pair for one b: coalesced v2f global read,
    // contiguous v2f LDS write.
    for (int i = tid; i < (D_IN / 2) * BTILE; i += 256) {
        const int pd = i & 63;             // d-pair index, d = 2*pd
        const int b  = i >> 6;
        v2f v = *(const v2f*)(x + (size_t)(b0 + b) * D_IN + 2 * pd);
        *(v2f*)(hbuf + (size_t)pd * PAIRW + 2 * b) = v;
    }
    __syncthreads();

    run_layer(W1 + (size_t)k * H_DIM * D_IN,  b1 + (size_t)k * H_DIM,
              D_IN,  D_IN,  hbuf, wave, lane);
    run_layer(W2 + (size_t)k * H_DIM * H_DIM, b2 + (size_t)k * H_DIM,
              H_DIM, H_DIM, hbuf, wave, lane);
    run_layer(W3 + (size_t)k * H_DIM * H_DIM, b3 + (size_t)k * H_DIM,
              H_DIM, H_DIM, hbuf, wave, lane);

    // Layer 4: out[b] = dot(W4[k], h[:, b]) + b4[k]  (out_features = 1)
    const float* W4k = W4 + (size_t)k * H_DIM;
    const int bcol = tid & 63;
    const int part = tid >> 6;             // 0..3, each sums 128 of 512 rows
    float s = 0.f;
    #pragma unroll 4
    for (int d = part * 128; d < part * 128 + 128; d += 2) {
        v2f w = *(const v2f*)(W4k + d);
        v2f h = *(const v2f*)(hbuf + (size_t)(d >> 1) * PAIRW + 2 * bcol);
        s += w.x * h.x + w.y * h.y;
    }
    red[tid] = s;
    __syncthreads();
    if (tid < 64) {
        const float v = red[tid] + red[tid + 64] + red[tid + 128] + red[tid + 192] + b4[k];
        // output layout [B, 1, K]
        out[(size_t)(b0 + tid) * NUM_K + k] = v;
    }
}

extern "C" void kernel_launch(void* const* d_in, const int* in_sizes, int n_in,
                              void* d_out, int out_size, void* d_ws, size_t ws_size,
                              hipStream_t stream) {
    const float* x  = (const float*)d_in[0];
    const float* W1 = (const float*)d_in[1];
    const float* b1 = (const float*)d_in[2];
    const float* W2 = (const float*)d_in[3];
    const float* b2 = (const float*)d_in[4];
    const float* W3 = (const float*)d_in[5];
    const float* b3 = (const float*)d_in[6];
    const float* W4 = (const float*)d_in[7];
    const float* b4 = (const float*)d_in[8];
    float* out = (float*)d_out;

    dim3 grid(BATCH / BTILE, NUM_K);       // (32, 64) = 2048 workgroups
    dim3 block(256);
    parallel_mlp_kernel<<<grid, block, 0, stream>>>(x, W1, b1, W2, b2, W3, b3, W4, b4, out);
}